// PHGCN_13975823581431
// MI455X (gfx1250) — compile-verified
//
#include <hip/hip_runtime.h>
#include <hip/hip_bf16.h>

typedef float v2f __attribute__((ext_vector_type(2)));
typedef float v8f __attribute__((ext_vector_type(8)));

#define TPB 256

// ---------- degree / norm ----------
__global__ void k_init_deg(float* deg, int N) {
    int i = blockIdx.x * TPB + threadIdx.x;
    if (i < N) deg[i] = 1.0f;           // self-loop contributes 1 to every node
}

__global__ void k_deg_accum(const int* __restrict__ dst, float* deg, int E) {
    int e = blockIdx.x * TPB + threadIdx.x;
    if (e < E) atomicAdd(&deg[dst[e]], 1.0f);
}

__global__ void k_dinv(float* deg, int N) {
    int i = blockIdx.x * TPB + threadIdx.x;
    if (i < N) {
        float d = deg[i];
        deg[i] = (d > 0.0f) ? rsqrtf(d) : 0.0f;
    }
}

// ---------- fused head matrices ----------
__global__ void k_fuse_heads(const float* __restrict__ L0, const float* __restrict__ b0,
                             const float* __restrict__ L1, const float* __restrict__ b1,
                             const float* __restrict__ Wo, const float* __restrict__ bo,
                             float* W0f, float* W1f, float* cb) {
    for (int i = threadIdx.x; i < 64 * 40; i += TPB) {
        int k = i / 40, c = i % 40;
        float s0 = 0.0f, s1 = 0.0f;
        for (int j = 0; j < 64; ++j) {
            s0 += L0[k * 64 + j] * Wo[j * 40 + c];
            s1 += L1[k * 64 + j] * Wo[(64 + j) * 40 + c];
        }
        W0f[i] = s0;
        W1f[i] = s1;
    }
    if (threadIdx.x < 40) {
        int c = threadIdx.x;
        float s = bo[c];
        for (int j = 0; j < 64; ++j)
            s += b0[j] * Wo[j * 40 + c] + b1[j] * Wo[(64 + j) * 40 + c];
        cb[c] = s;
    }
}

// ---------- WMMA GEMM: D[N x 64] = A[N x K] @ W[K x 64], K in {64,128} ----------
// Wave computes a 32-row x 64-col slab: 2 A fragments x 4 B col-tiles = 8 accumulators.
// W is staged in LDS pair-interleaved: sW[(k>>1)*128 + 2*col + (k&1)] so that each
// B fragment {W[kb][col], W[kb+1][col]} is one aligned 8-byte LDS load.
__global__ __launch_bounds__(128)
void k_gemm_n64_wmma(const float* __restrict__ A, const float* __restrict__ W,
                     float* __restrict__ D, int N, int K) {
    __shared__ float sW[128 * 64];      // 32 KB max (K<=128)
    int tid = threadIdx.x;
    for (int i = tid; i < K * 64; i += 128) {
        int k = i >> 6, col = i & 63;
        sW[(k >> 1) * 128 + 2 * col + (k & 1)] = W[i];
    }
    __syncthreads();

    int wave = tid >> 5;
    int lane = tid & 31;
    int half = lane >> 4;               // 0: K {0,1}, rows 0-7 ; 1: K {2,3}, rows 8-15
    int l16  = lane & 15;

    int rbase = blockIdx.x * 128 + wave * 32;
    int ar0 = rbase + l16;              // A fragment row-tile 0 (M = lane%16)
    int ar1 = rbase + 16 + l16;         // A fragment row-tile 1
    long ab0 = (long)((ar0 < N) ? ar0 : (N - 1)) * K;
    long ab1 = (long)((ar1 < N) ? ar1 : (N - 1)) * K;

    v8f acc00 = {}, acc01 = {}, acc02 = {}, acc03 = {};
    v8f acc10 = {}, acc11 = {}, acc12 = {}, acc13 = {};

    for (int k = 0; k < K; k += 4) {
        int kb = k + 2 * half;
        v2f a0 = *(const v2f*)(A + ab0 + kb);
        v2f a1 = *(const v2f*)(A + ab1 + kb);
        const float* bp = &sW[(kb >> 1) * 128 + 2 * l16];
        v2f b0 = *(const v2f*)(bp + 0);
        v2f b1 = *(const v2f*)(bp + 32);
        v2f b2 = *(const v2f*)(bp + 64);
        v2f b3 = *(const v2f*)(bp + 96);
        acc00 = __builtin_amdgcn_wmma_f32_16x16x4_f32(false, a0, false, b0, (short)0, acc00, false, false);
        acc01 = __builtin_amdgcn_wmma_f32_16x16x4_f32(false, a0, false, b1, (short)0, acc01, false, false);
        acc02 = __builtin_amdgcn_wmma_f32_16x16x4_f32(false, a0, false, b2, (short)0, acc02, false, false);
        acc03 = __builtin_amdgcn_wmma_f32_16x16x4_f32(false, a0, false, b3, (short)0, acc03, false, false);
        acc10 = __builtin_amdgcn_wmma_f32_16x16x4_f32(false, a1, false, b0, (short)0, acc10, false, false);
        acc11 = __builtin_amdgcn_wmma_f32_16x16x4_f32(false, a1, false, b1, (short)0, acc11, false, false);
        acc12 = __builtin_amdgcn_wmma_f32_16x16x4_f32(false, a1, false, b2, (short)0, acc12, false, false);
        acc13 = __builtin_amdgcn_wmma_f32_16x16x4_f32(false, a1, false, b3, (short)0, acc13, false, false);
    }

    // C/D layout: lane holds column n = lane%16, rows m = (lane/16)*8 + r
    int m0 = rbase + half * 8;          // row-tile 0
    int m1 = rbase + 16 + half * 8;     // row-tile 1
    for (int r = 0; r < 8; ++r) {
        int rr = m0 + r;
        if (rr < N) {
            float* dp = D + (long)rr * 64 + l16;
            dp[0]  = acc00[r];
            dp[16] = acc01[r];
            dp[32] = acc02[r];
            dp[48] = acc03[r];
        }
        int ss = m1 + r;
        if (ss < N) {
            float* dp = D + (long)ss * 64 + l16;
            dp[0]  = acc10[r];
            dp[16] = acc11[r];
            dp[32] = acc12[r];
            dp[48] = acc13[r];
        }
    }
}

// ---------- edge scatter: B[dst] += T[src] * dinv[src]*dinv[dst], 64 feats/edge ----------
__global__ void k_scatter(const float* __restrict__ T, const int* __restrict__ src,
                          const int* __restrict__ dst, const float* __restrict__ dinv,
                          float* Bagg, int E) {
    long gid = (long)blockIdx.x * TPB + threadIdx.x;
    int e = (int)(gid >> 6);
    int f = (int)(gid & 63);
    if (e < E) {
        int s = src[e], d = dst[e];
        float nr = dinv[s] * dinv[d];
        atomicAdd(&Bagg[(long)d * 64 + f], T[(long)s * 64 + f] * nr);
    }
}

// ---------- epilogue: H = elu(Bagg + dinv^2 * T + bias) (self-loop fused) ----------
__global__ void k_bias_selfloop_elu(const float* __restrict__ Bagg, const float* __restrict__ T,
                                    const float* __restrict__ dinv, const float* __restrict__ bias,
                                    float* H, int N) {
    long gid = (long)blockIdx.x * TPB + threadIdx.x;
    if (gid < (long)N * 64) {
        int n = (int)(gid >> 6);
        int f = (int)(gid & 63);
        float di = dinv[n];
        float v = Bagg[gid] + di * di * T[gid] + bias[f];
        H[gid] = (v > 0.0f) ? v : (expf(v) - 1.0f);
    }
}

// ---------- output: out[n,c] = cbias[c] + h0[n]·W0f[:,c] + h1[n]·W1f[:,c] ----------
__global__ void k_out(const float* __restrict__ H0, const float* __restrict__ H1,
                      const float* __restrict__ W0f, const float* __restrict__ W1f,
                      const float* __restrict__ cb, float* __restrict__ out, int N) {
    __shared__ float sw[64 * 40 * 2 + 40];
    for (int i = threadIdx.x; i < 64 * 40; i += TPB) {
        sw[i] = W0f[i];
        sw[2560 + i] = W1f[i];
    }
    if (threadIdx.x < 40) sw[5120 + threadIdx.x] = cb[threadIdx.x];
    __syncthreads();

    long gid = (long)blockIdx.x * TPB + threadIdx.x;
    if (gid < (long)N * 40) {
        int n = (int)(gid / 40);
        int c = (int)(gid % 40);
        const float* h0 = H0 + (long)n * 64;
        const float* h1 = H1 + (long)n * 64;
        float s = sw[5120 + c];
        for (int k = 0; k < 64; ++k)
            s += h0[k] * sw[k * 40 + c] + h1[k] * sw[2560 + k * 40 + c];
        out[gid] = s;
    }
}

extern "C" void kernel_launch(void* const* d_in, const int* in_sizes, int n_in,
                              void* d_out, int out_size, void* d_ws, size_t ws_size,
                              hipStream_t stream) {
    const float* x       = (const float*)d_in[0];
    const int*   ei      = (const int*)d_in[1];
    const float* conv0_w = (const float*)d_in[2];
    const float* conv0_b = (const float*)d_in[3];
    const float* lin0_w  = (const float*)d_in[4];
    const float* lin0_b  = (const float*)d_in[5];
    const float* conv1_w = (const float*)d_in[6];
    const float* conv1_b = (const float*)d_in[7];
    const float* lin1_w  = (const float*)d_in[8];
    const float* lin1_b  = (const float*)d_in[9];
    const float* out_w   = (const float*)d_in[10];
    const float* out_b   = (const float*)d_in[11];
    float* out = (float*)d_out;

    int N = in_sizes[0] / 128;
    int E = in_sizes[1] / 2;
    const int* src = ei;
    const int* dst = ei + E;

    size_t NH = (size_t)N * 64;
    float* ws   = (float*)d_ws;
    float* dinv = ws;
    float* A    = ws + (((size_t)N + 255) / 256) * 256;   // t = h @ W buffer
    float* B    = A + NH;                                  // aggregate, then h1 in-place
    float* H0   = B + NH;                                  // h0 (needed by output kernel)
    float* W0f  = H0 + NH;
    float* W1f  = W0f + 64 * 40;
    float* cb   = W1f + 64 * 40;

    int gN   = (N + TPB - 1) / TPB;
    int gE   = (E + TPB - 1) / TPB;
    int gNH  = (int)((NH + TPB - 1) / TPB);
    int gSc  = (int)(((size_t)E * 64 + TPB - 1) / TPB);
    int gOut = (int)(((size_t)N * 40 + TPB - 1) / TPB);
    int gGemm = (N + 127) / 128;

    // normalization: deg (with self-loop) -> dinv
    k_init_deg<<<gN, TPB, 0, stream>>>(dinv, N);
    k_deg_accum<<<gE, TPB, 0, stream>>>(dst, dinv, E);
    k_dinv<<<gN, TPB, 0, stream>>>(dinv, N);

    // fused head matrices (tiny)
    k_fuse_heads<<<1, TPB, 0, stream>>>(lin0_w, lin0_b, lin1_w, lin1_b, out_w, out_b,
                                        W0f, W1f, cb);

    // ---- layer 0 ----
    k_gemm_n64_wmma<<<gGemm, 128, 0, stream>>>(x, conv0_w, A, N, 128);
    hipMemsetAsync(B, 0, NH * sizeof(float), stream);
    k_scatter<<<gSc, TPB, 0, stream>>>(A, src, dst, dinv, B, E);
    k_bias_selfloop_elu<<<gNH, TPB, 0, stream>>>(B, A, dinv, conv0_b, H0, N);

    // ---- layer 1 ----
    k_gemm_n64_wmma<<<gGemm, 128, 0, stream>>>(H0, conv1_w, A, N, 64);
    hipMemsetAsync(B, 0, NH * sizeof(float), stream);
    k_scatter<<<gSc, TPB, 0, stream>>>(A, src, dst, dinv, B, E);
    k_bias_selfloop_elu<<<gNH, TPB, 0, stream>>>(B, A, dinv, conv1_b, B, N); // h1 in-place

    // ---- output projection with fused heads ----
    k_out<<<gOut, TPB, 0, stream>>>(H0, B, W0f, W1f, cb, out, N);
}